// TracerModel_38972533244646
// MI455X (gfx1250) — compile-verified
//
#include <hip/hip_runtime.h>
#include <stdint.h>

// ---------------------------------------------------------------------------
// Cart-pole RK4 step, B = 4.19M independent states.
// Memory-bound streaming kernel: 36 B/element -> ~151 MB @ 23.3 TB/s ~ 6.5 us.
// CDNA5 path used: GLOBAL_LOAD_ASYNC_TO_LDS_B128 double-buffered pipeline
// (ASYNCcnt), hardware sin/cos/rcp to stay under the HBM roofline.
// ---------------------------------------------------------------------------

#define TPB  256
#define EPT  4
#define TILE (TPB * EPT)   // 1024 elements per tile, 4 KB per input array

// One lane moves 16 bytes global -> LDS asynchronously (tracked by ASYNCcnt).
__device__ __forceinline__ void async_ld16(const float* g, const void* l) {
    unsigned lds = (unsigned)(uintptr_t)l;              // low 32 bits of a
    unsigned long long ga = (unsigned long long)(uintptr_t)g; // generic LDS addr
    asm volatile("global_load_async_to_lds_b128 %0, %1, off"
                 :: "v"(lds), "v"(ga) : "memory");
}

struct P {
    float b, d, M, Ieff, ml, m2gl, IeffM;
};

// Coupled EOM derivative; only (v, theta, omega) feed back.
__device__ __forceinline__ void deriv(float F, float v_, float th, float om,
                                      const P& p, float& xdd, float& thdd) {
    float s = __sinf(th);                 // v_sin_f32 (native)
    float c = __cosf(th);                 // v_cos_f32 (native)
    float A   = F - p.b * v_ + p.ml * om * om * s;
    float Bq  = p.m2gl * s - p.d * om;
    float mlc = p.ml * c;
    float D   = p.IeffM - mlc * mlc;
    float r   = __builtin_amdgcn_rcpf(D); // v_rcp_f32
    r = r * (2.0f - D * r);               // one Newton step -> ~0.5 ulp
    xdd  = (p.Ieff * A - mlc * Bq) * r;
    thdd = (p.M * Bq   - mlc * A ) * r;
}

__device__ __forceinline__ float4 rk4_step(float x, float v, float th, float om,
                                           float u, const P& p) {
    const float DT   = 0.019999999552965164f;
    const float half = (float)(0.019999999552965164 / 2.0);
    const float dt6  = (float)(0.019999999552965164 / 6.0);
    const float F = 100.0f * u;

    float xdd1, thdd1, xdd2, thdd2, xdd3, thdd3, xdd4, thdd4;
    deriv(F, v, th, om, p, xdd1, thdd1);
    float v1 = v + half * xdd1, om1 = om + half * thdd1;
    deriv(F, v1, th + half * om, om1, p, xdd2, thdd2);
    float v2 = v + half * xdd2, om2 = om + half * thdd2;
    deriv(F, v2, th + half * om1, om2, p, xdd3, thdd3);
    float v3 = v + DT * xdd3, om3 = om + DT * thdd3;
    deriv(F, v3, th + DT * om2, om3, p, xdd4, thdd4);

    float4 r;
    r.x = x  + dt6 * (v     + 2.0f * v1    + 2.0f * v2    + v3);
    r.y = v  + dt6 * (xdd1  + 2.0f * xdd2  + 2.0f * xdd3  + xdd4);
    r.z = th + dt6 * (om    + 2.0f * om1   + 2.0f * om2   + om3);
    r.w = om + dt6 * (thdd1 + 2.0f * thdd2 + 2.0f * thdd3 + thdd4);
    return r;
}

__global__ __launch_bounds__(TPB) void cartpole_rk4_kernel(
    const float* __restrict__ gx,  const float* __restrict__ gv,
    const float* __restrict__ gth, const float* __restrict__ gom,
    const float* __restrict__ gu,
    const float* __restrict__ pI,  const float* __restrict__ pb,
    const float* __restrict__ pd,  const float* __restrict__ pl,
    const float* __restrict__ pm1, const float* __restrict__ pm2,
    float4* __restrict__ out, int n)
{
    // Double-buffered staging: 2 buffers x 5 arrays x 256 lanes x 16B = 40 KB.
    __shared__ float4 sm[2][5][TPB];
    const int tid = threadIdx.x;
    const int ntiles = n / TILE;

    // Loop-invariant parameter algebra (single-element arrays, L2-resident).
    const float I  = *pI, b = *pb, d = *pd, l = *pl, m1 = *pm1, m2 = *pm2;
    P p;
    p.b = b; p.d = d;
    p.M = m1 + m2;
    p.ml = m2 * l;
    p.Ieff = I + p.ml * l;
    p.m2gl = p.ml * 9.8100004196167f;
    p.IeffM = p.Ieff * p.M;

#define STAGE(pbuf, t)                                          \
    do {                                                        \
        int base_ = (t) * TILE + tid * EPT;                     \
        async_ld16(gx  + base_, &sm[(pbuf)][0][tid]);           \
        async_ld16(gv  + base_, &sm[(pbuf)][1][tid]);           \
        async_ld16(gth + base_, &sm[(pbuf)][2][tid]);           \
        async_ld16(gom + base_, &sm[(pbuf)][3][tid]);           \
        async_ld16(gu  + base_, &sm[(pbuf)][4][tid]);           \
    } while (0)

    int tile = blockIdx.x;
    if (tile < ntiles) {
        STAGE(0, tile);                       // prime the pipeline
        int par = 0;
        for (; tile < ntiles; tile += gridDim.x) {
            const int nt = tile + gridDim.x;
            if (nt < ntiles) {
                STAGE(par ^ 1, nt);           // prefetch next tile
                // 5 new loads in flight; loads complete in order, so cnt<=5
                // guarantees the current buffer's 5 loads have landed in LDS.
                asm volatile("s_wait_asynccnt 0x5" ::: "memory");
            } else {
                asm volatile("s_wait_asynccnt 0x0" ::: "memory");
            }
            // Each lane consumes exactly the 16B slices it staged itself:
            // ASYNCcnt wait is sufficient, no workgroup barrier needed.
            const float4 X  = sm[par][0][tid];
            const float4 V  = sm[par][1][tid];
            const float4 TH = sm[par][2][tid];
            const float4 OM = sm[par][3][tid];
            const float4 U  = sm[par][4][tid];

            const int obase = tile * TILE + tid * EPT;  // 64B contiguous rows
            out[obase + 0] = rk4_step(X.x, V.x, TH.x, OM.x, U.x, p);
            out[obase + 1] = rk4_step(X.y, V.y, TH.y, OM.y, U.y, p);
            out[obase + 2] = rk4_step(X.z, V.z, TH.z, OM.z, U.z, p);
            out[obase + 3] = rk4_step(X.w, V.w, TH.w, OM.w, U.w, p);
            par ^= 1;
        }
    }

    // Tail (n not a multiple of TILE; empty for B = 4,194,304).
    const int rem_start = ntiles * TILE;
    for (int i = rem_start + (int)blockIdx.x * TPB + tid; i < n;
         i += (int)gridDim.x * TPB) {
        out[i] = rk4_step(gx[i], gv[i], gth[i], gom[i], gu[i], p);
    }
#undef STAGE
}

extern "C" void kernel_launch(void* const* d_in, const int* in_sizes, int n_in,
                              void* d_out, int out_size, void* d_ws, size_t ws_size,
                              hipStream_t stream) {
    (void)n_in; (void)out_size; (void)d_ws; (void)ws_size;
    const float* gx  = (const float*)d_in[0];
    const float* gv  = (const float*)d_in[1];
    const float* gth = (const float*)d_in[2];
    const float* gom = (const float*)d_in[3];
    const float* gu  = (const float*)d_in[4];
    const float* pI  = (const float*)d_in[5];
    const float* pb  = (const float*)d_in[6];
    const float* pd  = (const float*)d_in[7];
    const float* pl  = (const float*)d_in[8];
    const float* pm1 = (const float*)d_in[9];
    const float* pm2 = (const float*)d_in[10];
    float4* out = (float4*)d_out;

    const int n = in_sizes[0];
    const int ntiles = n / TILE;
    // Persistent blocks: each processes >=1 tile; cap so blocks grid-stride
    // over several tiles and the double-buffer pipeline has work to overlap.
    int grid = ntiles < 1 ? 1 : (ntiles < 1024 ? ntiles : 1024);

    cartpole_rk4_kernel<<<grid, TPB, 0, stream>>>(
        gx, gv, gth, gom, gu, pI, pb, pd, pl, pm1, pm2, out, n);
}